// SpatialGraph_70738111365774
// MI455X (gfx1250) — compile-verified
//
#include <hip/hip_runtime.h>
#include <hip/hip_bf16.h>
#include <math.h>

// Problem constants (match reference)
#define BATCH 64
#define CC    80    // num classes / tokens
#define DD    256   // hidden dim
#define NBOX  32

typedef float v2f __attribute__((ext_vector_type(2)));
typedef float v8f __attribute__((ext_vector_type(8)));

#define WMMA_F32(A, B, ACC) \
    __builtin_amdgcn_wmma_f32_16x16x4_f32(false, (A), false, (B), (short)0, (ACC), false, false)

// ---------------------------------------------------------------------------
// Kernel 1: spatial affinity from bboxes (bitmask rasterize + popcount L1)
// One block per batch image.
// ---------------------------------------------------------------------------
__global__ void __launch_bounds__(256) affinity_kernel(
    const float* __restrict__ bboxes,      // [B, N, 4]
    const int*   __restrict__ bclass,      // [B, N]
    float*       __restrict__ aff)         // [B, C, C]
{
    __shared__ unsigned m20[CC][13];   // 400 bits per class, scale 20
    __shared__ unsigned m10[CC][4];    // 100 bits per class, scale 10
    __shared__ float    sA[CC * CC];
    __shared__ float    sInvDen[CC];

    const int b   = blockIdx.x;
    const int tid = threadIdx.x;

    for (int i = tid; i < CC * 13; i += 256) (&m20[0][0])[i] = 0u;
    for (int i = tid; i < CC * 4;  i += 256) (&m10[0][0])[i] = 0u;
    __syncthreads();

    if (tid < NBOX) {
        const float* bb = bboxes + (size_t)(b * NBOX + tid) * 4;
        float bx = bb[0], by = bb[1], bw = bb[2], bh = bb[3];
        float x1 = fminf(fmaxf(bx,      0.f), 223.f);
        float y1 = fminf(fmaxf(by,      0.f), 223.f);
        float x2 = fminf(fmaxf(bx + bw, 0.f), 223.f);
        float y2 = fminf(fmaxf(by + bh, 0.f), 223.f);
        int c = bclass[b * NBOX + tid];
        for (int si = 0; si < 2; ++si) {
            const int   S  = si ? 20 : 10;
            const float fs = (float)S;
            int gx1 = (int)floorf(x1 * fs / 224.f);
            int gy1 = (int)floorf(y1 * fs / 224.f);
            int gx2 = (int)floorf(x2 * fs / 224.f);
            int gy2 = (int)floorf(y2 * fs / 224.f);
            unsigned* base = si ? &m20[c][0] : &m10[c][0];
            for (int yy = gy1; yy <= gy2; ++yy)
                for (int xx = gx1; xx <= gx2; ++xx) {
                    int bit = yy * S + xx;
                    atomicOr(&base[bit >> 5], 1u << (bit & 31));
                }
        }
    }
    __syncthreads();

    // aff[i][j] = 0.5*(1 - l1_10/100) + 0.5*(1 - l1_20/400)
    for (int idx = tid; idx < CC * CC; idx += 256) {
        int i = idx / CC, j = idx % CC;
        int l20 = 0;
        #pragma unroll
        for (int w = 0; w < 13; ++w) l20 += __popc(m20[i][w] ^ m20[j][w]);
        int l10 = 0;
        #pragma unroll
        for (int w = 0; w < 4; ++w)  l10 += __popc(m10[i][w] ^ m10[j][w]);
        sA[idx] = 0.5f * (1.f - (float)l10 * (1.f / 100.f))
                + 0.5f * (1.f - (float)l20 * (1.f / 400.f));
    }
    __syncthreads();

    if (tid < CC) {
        float s = 0.f;
        for (int j = 0; j < CC; ++j) s += fabsf(sA[tid * CC + j]);
        sInvDen[tid] = 1.0f / fmaxf(s, 1e-12f);
    }
    __syncthreads();

    for (int idx = tid; idx < CC * CC; idx += 256)
        aff[(size_t)b * CC * CC + idx] = sA[idx] * sInvDen[idx / CC];
}

// ---------------------------------------------------------------------------
// GEMM core: one wave computes a 32x64 strip of C = A@B + bias.
// Per k-step: 4 B-fragments (batched loads) + 2 A-fragments feed 8 wmma.
// ---------------------------------------------------------------------------
__device__ __forceinline__ void gemm_strip_32x64(
    const float* __restrict__ A, const float* __restrict__ B,
    const float* __restrict__ bias, float* __restrict__ C,
    int N, int K, int mBase, int nBase)
{
    const int lane = threadIdx.x & 31;
    const int half = lane >> 4;        // selects K pair {0,1} vs {2,3}
    const int l    = lane & 15;

    const float* Arow0 = A + (size_t)(mBase + l) * K + 2 * half;
    const float* Arow1 = Arow0 + (size_t)16 * K;
    const float* Bp    = B + (size_t)(2 * half) * N + nBase + l;

    v8f acc[2][4] = {};
    for (int k = 0; k < K; k += 4) {
        const float* Bk = Bp + (size_t)k * N;
        v2f bfr[4];
        #pragma unroll
        for (int nb = 0; nb < 4; ++nb) {    // batched: loads issue as a clause
            bfr[nb].x = Bk[nb * 16];
            bfr[nb].y = Bk[nb * 16 + (size_t)N];
        }
        v2f a0, a1;
        a0.x = Arow0[k]; a0.y = Arow0[k + 1];
        a1.x = Arow1[k]; a1.y = Arow1[k + 1];
        #pragma unroll
        for (int nb = 0; nb < 4; ++nb) {
            acc[0][nb] = WMMA_F32(a0, bfr[nb], acc[0][nb]);
            acc[1][nb] = WMMA_F32(a1, bfr[nb], acc[1][nb]);
        }
    }
    #pragma unroll
    for (int mt = 0; mt < 2; ++mt) {
        #pragma unroll
        for (int nb = 0; nb < 4; ++nb) {
            const int col = nBase + nb * 16 + l;
            const float bv = bias[col];
            #pragma unroll
            for (int g = 0; g < 8; ++g) {
                int row = mBase + mt * 16 + g + 8 * half;
                C[(size_t)row * N + col] = acc[mt][nb][g] + bv;
            }
        }
    }
}

// Merged q/k/v projection: blockIdx.z selects which weight/bias/output.
__global__ void __launch_bounds__(32) qkv_gemm(
    const float* __restrict__ x,
    const float* __restrict__ Wq, const float* __restrict__ bq, float* __restrict__ q,
    const float* __restrict__ Wk, const float* __restrict__ bk, float* __restrict__ kk,
    const float* __restrict__ Wv, const float* __restrict__ bv, float* __restrict__ vv)
{
    const float* W; const float* bias; float* O;
    if (blockIdx.z == 0)      { W = Wq; bias = bq; O = q;  }
    else if (blockIdx.z == 1) { W = Wk; bias = bk; O = kk; }
    else                      { W = Wv; bias = bv; O = vv; }
    gemm_strip_32x64(x, W, bias, O, DD, DD, blockIdx.y * 32, blockIdx.x * 64);
}

// Single GEMM (used for the output projection).
__global__ void __launch_bounds__(32) wmma_gemm_bias(
    const float* __restrict__ A, const float* __restrict__ B,
    const float* __restrict__ bias, float* __restrict__ C)
{
    gemm_strip_32x64(A, B, bias, C, DD, DD, blockIdx.y * 32, blockIdx.x * 64);
}

// ---------------------------------------------------------------------------
// Kernel 3: attention per batch.  scores = (q k^T / 16) * aff ; softmax ;
// out = attn @ v.  One block (8 waves) per batch. C=80 -> 5 tiles of 16.
// ---------------------------------------------------------------------------
__global__ void __launch_bounds__(256) attn_kernel(
    const float* __restrict__ q, const float* __restrict__ k,
    const float* __restrict__ v, const float* __restrict__ aff,
    float* __restrict__ out)
{
    __shared__ float sS[CC * CC];   // scores -> attn, 25.6 KB

    const int b    = blockIdx.x;
    const int tid  = threadIdx.x;
    const int wave = tid >> 5;
    const int lane = tid & 31;
    const int half = lane >> 4;
    const int l    = lane & 15;

    const float* qb = q + (size_t)b * CC * DD;
    const float* kb = k + (size_t)b * CC * DD;
    const float* vb = v + (size_t)b * CC * DD;
    const float* ab = aff + (size_t)b * CC * CC;

    // ---- stage 1: scores[m][n] = (q_m . k_n)/16 * aff[m][n] ---------------
    // Waves 0..4 each own one 16-row strip; one q-fragment feeds 5 wmma.
    if (wave < 5) {
        const int mBase = wave * 16;
        const float* qrow = qb + (size_t)(mBase + l) * DD + 2 * half;
        const float* kcol = kb + (size_t)l * DD + 2 * half;
        v8f acc[5] = {};
        for (int k0 = 0; k0 < DD; k0 += 4) {
            v2f bfr[5];
            #pragma unroll
            for (int tn = 0; tn < 5; ++tn) {  // B[kk][n] = k[n][kk]
                bfr[tn].x = kcol[(size_t)(tn * 16) * DD + k0];
                bfr[tn].y = kcol[(size_t)(tn * 16) * DD + k0 + 1];
            }
            v2f a;
            a.x = qrow[k0];
            a.y = qrow[k0 + 1];
            #pragma unroll
            for (int tn = 0; tn < 5; ++tn)
                acc[tn] = WMMA_F32(a, bfr[tn], acc[tn]);
        }
        #pragma unroll
        for (int tn = 0; tn < 5; ++tn) {
            #pragma unroll
            for (int g = 0; g < 8; ++g) {
                int row = mBase + g + 8 * half;
                int col = tn * 16 + l;
                sS[row * CC + col] = acc[tn][g] * 0.0625f * ab[row * CC + col];
            }
        }
    }
    __syncthreads();

    // ---- stage 2: row softmax --------------------------------------------
    if (tid < CC) {
        float mx = -INFINITY;
        for (int j = 0; j < CC; ++j) mx = fmaxf(mx, sS[tid * CC + j]);
        float s = 0.f;
        for (int j = 0; j < CC; ++j) {
            float e = expf(sS[tid * CC + j] - mx);
            sS[tid * CC + j] = e;
            s += e;
        }
        float inv = 1.0f / s;
        for (int j = 0; j < CC; ++j) sS[tid * CC + j] *= inv;
    }
    __syncthreads();

    // ---- stage 3: out = attn @ v  (M=80, N=256, K=80) --------------------
    // 20 items of 16x64; one attn-fragment (from LDS) feeds 4 wmma.
    for (int item = wave; item < 20; item += 8) {
        const int mBase = (item / 4) * 16;
        const int nBase = (item % 4) * 64;
        const float* arow = &sS[(mBase + l) * CC + 2 * half];
        const float* bcol = vb + (size_t)(2 * half) * DD + nBase + l;
        v8f acc[4] = {};
        for (int k0 = 0; k0 < CC; k0 += 4) {
            const float* Bk = bcol + (size_t)k0 * DD;
            v2f bfr[4];
            #pragma unroll
            for (int nb = 0; nb < 4; ++nb) {
                bfr[nb].x = Bk[nb * 16];
                bfr[nb].y = Bk[nb * 16 + DD];
            }
            v2f a;
            a.x = arow[k0];
            a.y = arow[k0 + 1];
            #pragma unroll
            for (int nb = 0; nb < 4; ++nb)
                acc[nb] = WMMA_F32(a, bfr[nb], acc[nb]);
        }
        #pragma unroll
        for (int nb = 0; nb < 4; ++nb) {
            #pragma unroll
            for (int g = 0; g < 8; ++g) {
                int row = mBase + g + 8 * half;
                int col = nBase + nb * 16 + l;
                out[(size_t)b * CC * DD + (size_t)row * DD + col] = acc[nb][g];
            }
        }
    }
}

// ---------------------------------------------------------------------------
extern "C" void kernel_launch(void* const* d_in, const int* in_sizes, int n_in,
                              void* d_out, int out_size, void* d_ws, size_t ws_size,
                              hipStream_t stream) {
    const float* x       = (const float*)d_in[0];   // [B,C,D]
    const float* bboxes  = (const float*)d_in[1];   // [B,N,4]
    const int*   bclass  = (const int*)  d_in[2];   // [B,N]
    const float* Wq      = (const float*)d_in[3];
    const float* bq      = (const float*)d_in[4];
    const float* Wk      = (const float*)d_in[5];
    const float* bk      = (const float*)d_in[6];
    const float* Wv      = (const float*)d_in[7];
    const float* bv      = (const float*)d_in[8];
    const float* Wo      = (const float*)d_in[9];
    const float* bo      = (const float*)d_in[10];
    float*       outp    = (float*)d_out;

    // workspace layout (floats)
    const size_t affN = (size_t)BATCH * CC * CC;     // 409,600
    const size_t matN = (size_t)BATCH * CC * DD;     // 1,310,720
    float* ws_aff = (float*)d_ws;
    float* ws_q   = ws_aff + affN;
    float* ws_k   = ws_q + matN;
    float* ws_v   = ws_k + matN;
    float* ws_ao  = ws_v + matN;                     // attention output

    const int M = BATCH * CC;                        // 5120 rows

    // 1) spatial affinity
    affinity_kernel<<<dim3(BATCH), dim3(256), 0, stream>>>(bboxes, bclass, ws_aff);

    // 2) merged q/k/v projections: [5120,256] @ [256,256] + bias, z = which
    dim3 gQKV(DD / 64, M / 32, 3);                   // (4, 160, 3)
    qkv_gemm<<<gQKV, dim3(32), 0, stream>>>(x, Wq, bq, ws_q,
                                            Wk, bk, ws_k,
                                            Wv, bv, ws_v);

    // 3) affinity-modulated attention
    attn_kernel<<<dim3(BATCH), dim3(256), 0, stream>>>(ws_q, ws_k, ws_v, ws_aff, ws_ao);

    // 4) output projection -> d_out
    dim3 gGemm(DD / 64, M / 32);                     // (4, 160)
    wmma_gemm_bias<<<gGemm, dim3(32), 0, stream>>>(ws_ao, Wo, bo, outp);
}